// Conv2DCaps_69063074119717
// MI455X (gfx1250) — compile-verified
//
#include <hip/hip_runtime.h>

// ---------------- problem constants ----------------
// B=2, H=W=24, CH_I=32, N_I=8, CH_J=32, N_J=8, K=3x3 (KSZ=9), VALID -> 22x22
// P = 22*22 = 484 output positions (padded to 496 = 31*16 in ub scratch),
// I = KSZ*CH_I = 288, J = CH_J*N_J = 256, Q = 24*24 = 576 input positions,
// j_all = 484*32 = 15488, j_add = 15200
#define NB    2
#define HI    24
#define HJ    22
#define NP    484      // 22*22 (logical rows)
#define NPP   496      // 31*16 (padded rows in ub scratch)
#define NQ    576      // 24*24
#define NI    288      // 9*32
#define NJC   256      // 32*8
#define BLOGN ((long)NB*NQ*32*288)    // b_log / c element count
#define UBN   ((long)NB*NPP*NI*NJC)   // padded u_hat element count

typedef float v2f __attribute__((ext_vector_type(2)));
typedef float v8f __attribute__((ext_vector_type(8)));

// ---------------- zero-init ----------------
__global__ void k_zero(float* __restrict__ p, long n) {
    long g = (long)blockIdx.x * blockDim.x + threadIdx.x;
    if (g < n) p[g] = 0.f;
}

// ---------------- u_hat GEMM via V_WMMA_F32_16X16X4_F32 ----------------
// For each (b, i): (484 x 8) @ (8 x 256).  One wave handles one 16-row M-tile
// across all 16 N-tiles; K=8 is two K=4 WMMA steps.
// A layout (ISA 7.12.2, 32-bit A 16x4): lane m (0..15) holds K=0,1 in v0,v1;
// lane m+16 holds K=2,3.  B (4x16): vgpr r holds row r (lanes 0-15) and row
// r+2 (lanes 16-31), col = lane&15.  D (16x16): vgpr r -> row r / r+8.
// ub is M-padded to 496 rows so D stores need no bounds checks: one base
// pointer + 8 constant-offset global_store_b32 (row stride 294912 B < 2^23).
__global__ __launch_bounds__(128) void k_uhat(const float* __restrict__ x,
                                              const float* __restrict__ w,
                                              float* __restrict__ ub) {
    const int T = NB * NI * 31;                  // 31 M-tiles of 16 rows
    int wave = blockIdx.x * (blockDim.x >> 5) + (threadIdx.x >> 5);
    if (wave >= T) return;
    int lane = threadIdx.x & 31;
    int mt = wave % 31;
    int t2 = wave / 31;
    int i  = t2 % NI;
    int b  = t2 / NI;
    int k  = i >> 5, ci = i & 31;                // kernel pos, input channel
    int kh = k / 3, kw = k % 3;

    int m   = lane & 15;
    int hiL = lane >> 4;                         // 0 or 1 (lane half)
    int kb2 = hiL * 2;                           // K base {0,2}
    int p   = mt * 16 + m;
    int pc  = p < (NP - 1) ? p : (NP - 1);       // clamp A loads for pad rows
    int ph  = pc / HJ, pw = pc % HJ;

    // A rows: pt[b,p,i,0..7] = x[b, ph+kh, pw+kw, ci, 0..7]
    const float* xrow = x + ((((long)b * HI + ph + kh) * HI + (pw + kw)) * 32 + ci) * 8;
    v2f a0, a1;
    a0.x = xrow[kb2];     a0.y = xrow[kb2 + 1];      // K-step 0 (m'=0..3)
    a1.x = xrow[4 + kb2]; a1.y = xrow[4 + kb2 + 1];  // K-step 1 (m'=4..7)

    int n = lane & 15;
    const float* wbase = w + (long)i * 2048;         // wr[i, m, j], 8x256

    // per-lane D store base: row (mt*16 + 8*hiL), col n; rows r apart by NI*NJC
    float* outp = ub + (((long)b * NPP + mt * 16 + 8 * hiL) * NI + i) * NJC + n;

    for (int nt = 0; nt < 16; ++nt) {
        int j0 = nt * 16;
        v2f b0, b1;
        b0.x = wbase[(kb2)     * 256 + j0 + n];
        b0.y = wbase[(kb2 + 1) * 256 + j0 + n];
        b1.x = wbase[(4 + kb2)     * 256 + j0 + n];
        b1.y = wbase[(4 + kb2 + 1) * 256 + j0 + n];

        v8f acc = {0.f, 0.f, 0.f, 0.f, 0.f, 0.f, 0.f, 0.f};
        acc = __builtin_amdgcn_wmma_f32_16x16x4_f32(false, a0, false, b0,
                                                    (short)0, acc, false, false);
        acc = __builtin_amdgcn_wmma_f32_16x16x4_f32(false, a1, false, b1,
                                                    (short)0, acc, false, false);
        // straight-line stores, constant offsets (no guards: ub is M-padded)
        float* o = outp + j0;
#pragma unroll
        for (int r = 0; r < 8; ++r)
            o[(long)r * NI * NJC] = acc[r];
    }
}

// ---------------- routing sum + squash ----------------
// round 0: s = sum_i ub.  rounds 1/2: s = sum_i coef * ub with
// coef(b,p,i=k*32+ci,cj) = c[b, q=(ph+kh,pw+kw), ci, n=8-k, cj].
__global__ __launch_bounds__(256) void k_route(const float* __restrict__ ub,
                                               const float* __restrict__ c,
                                               float* __restrict__ vout,
                                               int use_coef) {
    int b  = blockIdx.x / NP;
    int p  = blockIdx.x % NP;
    int ph = p / HJ, pw = p % HJ;
    int j  = threadIdx.x;
    int cj = j >> 3;
    const float* ubp = ub + (((long)b * NPP + p) * NI) * NJC + j;

    float s = 0.f;
    if (!use_coef) {
        for (int i = 0; i < NI; ++i) s += ubp[(long)i * NJC];
    } else {
        for (int k = 0; k < 9; ++k) {
            int kh = k / 3, kw = k % 3;
            int q = (ph + kh) * HI + (pw + kw);
            int nslot = 8 - k;
            const float* cp = c + (((long)b * NQ + q) * 32) * 288 + nslot * 32 + cj;
            const float* up = ubp + (long)k * 32 * NJC;
            for (int ci = 0; ci < 32; ++ci)
                s += cp[(long)ci * 288] * up[(long)ci * NJC];
        }
    }
    // squash over the 8-lane n_j group (threads share cj in groups of 8)
    float sq = s * s;
    sq += __shfl_xor(sq, 1, 8);
    sq += __shfl_xor(sq, 2, 8);
    sq += __shfl_xor(sq, 4, 8);
    float v = (sq / (1.f + sq)) * (s * rsqrtf(sq + 1e-7f));
    vout[((long)b * NP + p) * NJC + j] = v;
}

// ---------------- agreement -> b_log update (injective scatter == gather) ---
// b_log[b,q,ci,n,cj] += sum_nj v[b,p,cj,nj]*ub[b,p,i,cj,nj]
// with k=8-n, (ph,pw)=(qh-kh,qw-kw), i=k*32+ci; zero if p out of bounds.
__global__ __launch_bounds__(256) void k_agree(const float* __restrict__ ub,
                                               const float* __restrict__ v,
                                               float* __restrict__ blog) {
    long g = (long)blockIdx.x * blockDim.x + threadIdx.x;
    if (g >= BLOGN) return;
    int cj    = (int)(g & 31);
    int nslot = (int)((g >> 5) % 9);
    long t    = g / 288;
    int ci    = (int)(t & 31);
    long t2   = t >> 5;
    int q     = (int)(t2 % NQ);
    int b     = (int)(t2 / NQ);
    int k  = 8 - nslot;
    int kh = k / 3, kw = k % 3;
    int qh = q / HI, qw = q % HI;
    int ph = qh - kh, pw = qw - kw;
    if (ph < 0 || ph >= HJ || pw < 0 || pw >= HJ) return;
    int p = ph * HJ + pw;
    int i = k * 32 + ci;
    const float* up = ub + (((long)b * NPP + p) * NI + i) * NJC + cj * 8;
    const float* vp = v  + ((long)b * NP + p) * NJC + cj * 8;
    float a = 0.f;
#pragma unroll
    for (int nj = 0; nj < 8; ++nj) a += vp[nj] * up[nj];
    blog[g] += a;
}

// ---------------- coupling coefficients ----------------
// c = exp(8*b_log) / (sum_288 exp(8*b_log) + eps + j_add) * j_all, per (b,q,ci)
__global__ __launch_bounds__(256) void k_coef(const float* __restrict__ blog,
                                              float* __restrict__ c) {
    const int NW = NB * NQ * 32;
    int waveg = blockIdx.x * (blockDim.x >> 5) + (threadIdx.x >> 5);
    if (waveg >= NW) return;
    int lane = threadIdx.x & 31;
    const float* bp = blog + (long)waveg * 288;
    float e[9];
    float sum = 0.f;
#pragma unroll
    for (int t = 0; t < 9; ++t) {
        float ex = __expf(8.f * bp[t * 32 + lane]);
        e[t] = ex;
        sum += ex;
    }
#pragma unroll
    for (int off = 16; off > 0; off >>= 1) sum += __shfl_xor(sum, off, 32);
    float scale = 15488.f / (sum + 1e-7f + 15200.f);
    float* cp = c + (long)waveg * 288;
#pragma unroll
    for (int t = 0; t < 9; ++t) cp[t * 32 + lane] = e[t] * scale;
}

// ---------------- host-side launcher ----------------
extern "C" void kernel_launch(void* const* d_in, const int* in_sizes, int n_in,
                              void* d_out, int out_size, void* d_ws, size_t ws_size,
                              hipStream_t stream) {
    const float* x = (const float*)d_in[0];   // (2,24,24,32,8)
    const float* w = (const float*)d_in[1];   // (3,3,32,8,32,8)
    float* out = (float*)d_out;               // (2,22,22,32,8)

    float* ub   = (float*)d_ws;               // ~293 MB (M-padded)
    float* blog = ub + UBN;                   // 42.5 MB
    float* cbuf = blog + BLOGN;               // 42.5 MB
    float* vbuf = cbuf + BLOGN;               // ~1 MB

    // b_log = 0
    {
        int blocks = (int)((BLOGN + 255) / 256);
        k_zero<<<blocks, 256, 0, stream>>>(blog, BLOGN);
    }
    // u_hat
    {
        int T = NB * NI * 31;
        k_uhat<<<(T + 3) / 4, 128, 0, stream>>>(x, w, ub);
    }
    int agree_blocks = (int)((BLOGN + 255) / 256);
    int coef_blocks  = (NB * NQ * 32 + 7) / 8;

    // round 0
    k_route<<<NB * NP, 256, 0, stream>>>(ub, cbuf, vbuf, 0);
    k_agree<<<agree_blocks, 256, 0, stream>>>(ub, vbuf, blog);
    // round 1
    k_coef<<<coef_blocks, 256, 0, stream>>>(blog, cbuf);
    k_route<<<NB * NP, 256, 0, stream>>>(ub, cbuf, vbuf, 1);
    k_agree<<<agree_blocks, 256, 0, stream>>>(ub, vbuf, blog);
    // round 2 -> final v straight to output
    k_coef<<<coef_blocks, 256, 0, stream>>>(blog, cbuf);
    k_route<<<NB * NP, 256, 0, stream>>>(ub, cbuf, out, 1);
}